// HyperedgeMeanAggregator_78408922955822
// MI455X (gfx1250) — compile-verified
//
#include <hip/hip_runtime.h>
#include <hip/hip_bf16.h>

// CDNA5 / gfx1250 HyperedgeMeanAggregator
//  kernel 1: per-hyperedge lower_bound over sorted seg_ids -> start offsets (in d_ws)
//  kernel 2: one wave per 16 hyperedges; segment-sum via V_WMMA_F32_16X16X4_F32
//            (D[16 segs x 16 feats] += A[16x4 selection] * B[4 entries x 16 feats]),
//            8 feature chunks cover FEAT=128; then scale by 1/count and store.

typedef __attribute__((ext_vector_type(2))) float v2f;
typedef __attribute__((ext_vector_type(8))) float v8f;

#define FEAT 128

static __device__ __forceinline__ int imin(int a, int b) { return a < b ? a : b; }
static __device__ __forceinline__ int imax(int a, int b) { return a > b ? a : b; }

__global__ __launch_bounds__(256) void seg_starts_kernel(const int* __restrict__ seg,
                                                         int E, int H,
                                                         int* __restrict__ start) {
    int h = blockIdx.x * blockDim.x + threadIdx.x;
    if (h > H) return;
    // lower_bound: first index i with seg[i] >= h   (seg is sorted ascending)
    int lo = 0, hi = E;
    while (lo < hi) {
        int mid = (lo + hi) >> 1;
        if (seg[mid] < h) lo = mid + 1; else hi = mid;
    }
    start[h] = lo;
}

__global__ __launch_bounds__(128) void hyperedge_mean_wmma_kernel(
    const float* __restrict__ table,   // [NUM_NODES, 128]
    const int*   __restrict__ node,    // [E]
    const int*   __restrict__ seg,     // [E] sorted
    const int*   __restrict__ start,   // [H+1]
    float*       __restrict__ out,     // [H, 128]
    int E, int H) {
    // One wave (32 lanes) handles 16 consecutive hyperedges x all 128 features.
    const int wave = blockIdx.x * 4 + (threadIdx.x >> 5);   // uniform per wave
    const int h0 = wave * 16;
    if (h0 >= H) return;                                    // uniform exit

    const int lane = threadIdx.x & 31;
    const int m  = lane & 15;          // N column (and A row M)
    const int hi = lane >> 4;          // lane half selects K pair

    const int hTop     = imin(h0 + 16, H);
    const int segStart = start[h0];
    const int segEnd   = start[hTop];
    const int Eclamp   = E - 1;

    v8f acc[8] = {};                   // 8 feature chunks of 16 -> 128 feats

    for (int e0 = segStart; e0 < segEnd; e0 += 4) {
        const int k  = e0 + 2 * hi;                 // this lane's K=0 entry
        const int k0 = imin(k,     Eclamp);
        const int k1 = imin(k + 1, Eclamp);

        // Prefetch upcoming index data (global_prefetch_b8, near scope)
        __builtin_prefetch(seg  + imin(k + 64, Eclamp), 0, 3);
        __builtin_prefetch(node + imin(k + 64, Eclamp), 0, 3);

        const int s0 = seg[k0];
        const int s1 = seg[k1];
        const int n0 = node[k0];
        const int n1 = node[k1];

        // A: 16x4 selection matrix. A[M=m][K] = 1 iff entry K belongs to segment h0+m.
        // Branch-free predicates (bitwise &) keep the loads unconditional and
        // avoid EXEC-mask churn inside the WMMA loop.
        const int p0 = (int)(k     < segEnd) & (int)((s0 - h0) == m);
        const int p1 = (int)(k + 1 < segEnd) & (int)((s1 - h0) == m);
        v2f a;
        a.x = p0 ? 1.0f : 0.0f;
        a.y = p1 ? 1.0f : 0.0f;

        const float* r0 = table + (size_t)n0 * FEAT + m;
        const float* r1 = table + (size_t)n1 * FEAT + m;

#pragma unroll
        for (int fc = 0; fc < 8; ++fc) {
            // B: 4x16 gathered feature chunk. B[K][N=m] = embed[node[e0+K]][fc*16+m]
            v2f b;
            b.x = r0[fc * 16];
            b.y = r1[fc * 16];
            acc[fc] = __builtin_amdgcn_wmma_f32_16x16x4_f32(
                /*neg_a=*/false, a, /*neg_b=*/false, b,
                /*c_mod=*/(short)0, acc[fc],
                /*reuse_a=*/false, /*reuse_b=*/false);
        }
    }

    // Epilogue: divide by segment count and store. (Divergence OK: no more WMMA.)
#pragma unroll
    for (int v = 0; v < 8; ++v) {
        const int h = h0 + v + 8 * hi;              // D row M = v + 8*hi
        if (h >= H) continue;
        const int cnt = start[h + 1] - start[h];
        const float inv = 1.0f / (float)imax(cnt, 1);
#pragma unroll
        for (int fc = 0; fc < 8; ++fc) {
            out[(size_t)h * FEAT + fc * 16 + m] = acc[fc][v] * inv;
        }
    }
}

extern "C" void kernel_launch(void* const* d_in, const int* in_sizes, int n_in,
                              void* d_out, int out_size, void* d_ws, size_t ws_size,
                              hipStream_t stream) {
    const float* table = (const float*)d_in[0];   // embed_table [N,128] f32
    const int*   node  = (const int*)d_in[1];     // node_idx [E]
    const int*   seg   = (const int*)d_in[2];     // seg_ids  [E] sorted
    // d_in[3] = num_hyperedges scalar; H derived host-side from out_size instead.

    const int E = in_sizes[1];
    const int H = out_size / FEAT;

    int*   start = (int*)d_ws;                    // (H+1) ints of scratch
    float* out   = (float*)d_out;

    // 1) segment start offsets via binary search
    {
        int threads = 256;
        int blocks  = (H + 1 + threads - 1) / threads;
        seg_starts_kernel<<<blocks, threads, 0, stream>>>(seg, E, H, start);
    }
    // 2) WMMA segment-mean aggregation: one wave per 16 hyperedges, 4 waves/block
    {
        int waves  = (H + 15) / 16;
        int blocks = (waves + 3) / 4;
        hyperedge_mean_wmma_kernel<<<blocks, 128, 0, stream>>>(table, node, seg, start, out, E, H);
    }
}